// LaplaceGatingNetwork_25709674234436
// MI455X (gfx1250) — compile-verified
//
#include <hip/hip_runtime.h>
#include <hip/hip_bf16.h>
#include <math.h>

typedef __attribute__((ext_vector_type(4))) float f4;
typedef __attribute__((ext_vector_type(2))) float v2f;
typedef __attribute__((ext_vector_type(8))) float v8f;

// ---------------------------------------------------------------------------
// Stage 1: global-average-pool over spatial dims.
// One block per contiguous 16384-float run (b,c). Blocks:
//   [0, 4096)    -> F1
//   [4096, 8192) -> F2
//   [8192, 8240) -> fr
// Each block: 256 threads x 16 float4 non-temporal loads -> LDS tree reduce.
// This stage is the roofline: 540 MB read once @ 23.3 TB/s ~= 23 us.
// ---------------------------------------------------------------------------
__global__ __launch_bounds__(256)
void lgn_pool_kernel(const float* __restrict__ F1,
                     const float* __restrict__ F2,
                     const float* __restrict__ fr,
                     float* __restrict__ ws) {
    const int blk = blockIdx.x;
    const int tid = threadIdx.x;

    const float* src;
    if (blk < 4096)       src = F1 + (size_t)blk * 16384;
    else if (blk < 8192)  src = F2 + (size_t)(blk - 4096) * 16384;
    else                  src = fr + (size_t)(blk - 8192) * 16384;

    const f4* s4 = (const f4*)src;

    float sum = 0.0f;
#pragma unroll
    for (int i = 0; i < 16; ++i) {
        // streamed once; non-temporal to avoid cache thrash
        f4 v = __builtin_nontemporal_load(&s4[i * 256 + tid]);
        sum += (v.x + v.y) + (v.z + v.w);
    }

    __shared__ float red[256];
    red[tid] = sum;
    __syncthreads();
#pragma unroll
    for (int s = 128; s > 0; s >>= 1) {
        if (tid < s) red[tid] += red[tid + s];
        __syncthreads();
    }
    if (tid == 0) ws[blk] = red[0] * (1.0f / 16384.0f);
}

// ---------------------------------------------------------------------------
// Stage 2: single workgroup (8 waves).
//   frp = fr_pooled(16x3) @ W^T(3x256) via V_WMMA_F32_16X16X4_F32 (K padded 3->4)
//   then per-batch top-3 of -|frp - x| for both tensors + softmax(3).
// ws layout (floats): [0,4096) x1 means, [4096,8192) x2 means, [8192,8240) fr_pooled
// out layout (floats): [0,48) w1 | [48,96) i1 | [96,144) w2 | [144,192) i2
// ---------------------------------------------------------------------------
__global__ __launch_bounds__(256)
void lgn_gate_kernel(const float* __restrict__ ws,
                     const float* __restrict__ W,   // (256, 3) row-major
                     float* __restrict__ out) {
    __shared__ float frp_lds[16 * 256];

    const int tid  = threadIdx.x;
    const int lane = tid & 31;
    const int wave = tid >> 5;

    const float* x1  = ws;
    const float* x2  = ws + 4096;
    const float* frp = ws + 8192;   // (16,3)

    // A fragment: 16x4 f32. Lanes 0-15 carry K={0,1}, lanes 16-31 carry K={2,3}.
    // Only K=3 is padding; K={0,1,2} loads are always in-bounds, so load
    // unconditionally and select — no exec-mask divergence in the WMMA region.
    const int  am = lane & 15;
    const bool lo = (lane < 16);
    const int  kbase = lo ? 0 : 2;

    v2f a;
    a.x = frp[am * 3 + kbase];                 // K = 0 or 2, always valid
    const float a_k1 = frp[am * 3 + 1];        // K = 1, always valid
    a.y = lo ? a_k1 : 0.0f;                    // K = 3 -> zero pad (cndmask)

    // Each wave computes two 16x16 output tiles (32 channels).
#pragma unroll
    for (int t = 0; t < 2; ++t) {
        const int c0 = wave * 32 + t * 16;
        const int n  = c0 + (lane & 15);

        // B fragment: 4x16 f32, mirrored striping. B[k][n] = W[n][k].
        v2f b;
        b.x = W[n * 3 + kbase];                // K = 0 or 2, max index 767: valid
        const float b_k1 = W[n * 3 + 1];       // K = 1, max index 766: valid
        b.y = lo ? b_k1 : 0.0f;                // K = 3 -> zero pad

        v8f c = {};
        c = __builtin_amdgcn_wmma_f32_16x16x4_f32(
                /*neg_a=*/false, a, /*neg_b=*/false, b,
                /*c_mod=*/(short)0, c, /*reuse_a=*/false, /*reuse_b=*/false);

        // D layout: VGPR r, lanes 0-15 -> M=r, lanes 16-31 -> M=r+8; N = lane%16
#pragma unroll
        for (int r = 0; r < 8; ++r) {
            const int mm = r + ((lane >> 4) << 3);
            frp_lds[mm * 256 + n] = c[r];
        }
    }
    __syncthreads();

    // Top-3 + softmax. Threads 0..15: tensor 1 (batch=tid); 16..31: tensor 2.
    if (tid < 32) {
        const int    bch  = tid & 15;
        const float* x    = (tid < 16) ? x1 : x2;
        float*       wout = out + ((tid < 16) ? 0 : 96);
        float*       iout = wout + 48;

        float v0 = -3.4e38f, v1 = -3.4e38f, v2 = -3.4e38f;
        int   i0 = 0, i1 = 0, i2 = 0;
        for (int cch = 0; cch < 256; ++cch) {
            const float d = -fabsf(frp_lds[bch * 256 + cch] - x[bch * 256 + cch]);
            if (d > v0)      { v2 = v1; i2 = i1; v1 = v0; i1 = i0; v0 = d; i0 = cch; }
            else if (d > v1) { v2 = v1; i2 = i1; v1 = d;  i1 = cch; }
            else if (d > v2) { v2 = d;  i2 = cch; }
        }
        // softmax over the sorted-descending triple (max = v0)
        const float e0 = 1.0f;
        const float e1 = __expf(v1 - v0);
        const float e2 = __expf(v2 - v0);
        const float inv = 1.0f / (e0 + e1 + e2);

        wout[bch * 3 + 0] = e0 * inv;
        wout[bch * 3 + 1] = e1 * inv;
        wout[bch * 3 + 2] = e2 * inv;
        iout[bch * 3 + 0] = (float)i0;
        iout[bch * 3 + 1] = (float)i1;
        iout[bch * 3 + 2] = (float)i2;
    }
}

extern "C" void kernel_launch(void* const* d_in, const int* in_sizes, int n_in,
                              void* d_out, int out_size, void* d_ws, size_t ws_size,
                              hipStream_t stream) {
    const float* F1 = (const float*)d_in[0];   // (16,256,128,128)
    const float* F2 = (const float*)d_in[1];   // (16,256,128,128)
    const float* fr = (const float*)d_in[2];   // (16,3,128,128)
    const float* W  = (const float*)d_in[3];   // (256,3)

    float* ws  = (float*)d_ws;   // needs (8192+48)*4 = 32,960 bytes
    float* out = (float*)d_out;  // 192 floats

    lgn_pool_kernel<<<dim3(8240), dim3(256), 0, stream>>>(F1, F2, fr, ws);
    lgn_gate_kernel<<<dim3(1),    dim3(256), 0, stream>>>(ws, W, out);
}